// GumbelSinkhorn_57878979281316
// MI455X (gfx1250) — compile-verified
//
#include <hip/hip_runtime.h>
#include <hip/hip_bf16.h>
#include <math.h>

// Gumbel-Sinkhorn masked softmax iterations for MI455X (gfx1250, wave32).
//
// B=128, N=512, M=512 fp32. Memory-bound: ~2.7 GB total traffic across the
// 10 softmax passes (~115us at 23.3 TB/s HBM; 134MB working set fits in the
// 192MB L2, so replays are mostly L2-resident). Design: each pass reads the
// tensor once and writes it once, in place. The strided (over-agents) softmax
// stages a 512x32 fp32 tile in LDS (66KB of the 320KB WGP LDS) so all global
// traffic is 128B-coalesced. Column sums-of-exp are computed with
// V_WMMA_F32_16X16X4_F32 (ones^T x expTile), which is exact fp32 and uses the
// CDNA5 matrix pipe; max reductions use LDS / wave32 shuffles.

#define NROW 512
#define MCOL 512
#define TILE_STRIDE 33   // 512x33 fp32 LDS tile, padded against bank conflicts

typedef __attribute__((ext_vector_type(2))) float v2f;
typedef __attribute__((ext_vector_type(8))) float v8f;

// ---------------------------------------------------------------------------
// Row softmax (over tasks axis, contiguous): one wave32 per row.
// Lane holds 16 elements (4 x float4, coalesced 512B per wave access).
// In-place safe: a wave loads its entire row before storing.
// ---------------------------------------------------------------------------
__global__ __launch_bounds__(256) void row_softmax_kernel(
    const float* __restrict__ src, float* __restrict__ dst,
    const int* __restrict__ fa_num, const int* __restrict__ tn_num,
    float scale)
{
    const int t    = threadIdx.x;
    const int lane = t & 31;
    const int w    = t >> 5;
    const int row  = blockIdx.x * 8 + w;   // global row in [0, B*N)
    const int b    = row >> 9;             // / 512
    const int n    = row & 511;
    const int fa   = fa_num[b];
    const int tn   = tn_num[b];

    const float* srow = src + (size_t)row * MCOL;
    float*       drow = dst + (size_t)row * MCOL;

    // Coalesced load: 4 chunks of 128 floats, lane l takes float4 at l*16B.
    float4 v[4];
#pragma unroll
    for (int c = 0; c < 4; ++c)
        v[c] = reinterpret_cast<const float4*>(srow)[c * 32 + lane];

    float xv[16];
#pragma unroll
    for (int c = 0; c < 4; ++c) {
        xv[c * 4 + 0] = v[c].x * scale;
        xv[c * 4 + 1] = v[c].y * scale;
        xv[c * 4 + 2] = v[c].z * scale;
        xv[c * 4 + 3] = v[c].w * scale;
    }

    const bool rowvalid = (n < fa) && (tn > 0);

    // Masked max over valid columns.
    float m = -INFINITY;
#pragma unroll
    for (int i = 0; i < 16; ++i) {
        const int col = (i >> 2) * 128 + lane * 4 + (i & 3);
        if (col < tn) m = fmaxf(m, xv[i]);
    }
#pragma unroll
    for (int off = 16; off >= 1; off >>= 1)
        m = fmaxf(m, __shfl_xor(m, off, 32));

    // exp + masked sum (masked lanes contribute exactly 0, matching the
    // reference where exp(NEG - max) underflows to 0).
    float e[16];
    float s = 0.f;
#pragma unroll
    for (int i = 0; i < 16; ++i) {
        const int col = (i >> 2) * 128 + lane * 4 + (i & 3);
        const float ei = (col < tn) ? __expf(xv[i] - m) : 0.f;
        e[i] = ei;
        s += ei;
    }
#pragma unroll
    for (int off = 16; off >= 1; off >>= 1)
        s += __shfl_xor(s, off, 32);

    const float inv = 1.f / s;   // s >= 1 whenever rowvalid (max is attained)

    float4 o[4];
#pragma unroll
    for (int c = 0; c < 4; ++c) {
        o[c].x = rowvalid ? e[c * 4 + 0] * inv : 0.f;
        o[c].y = rowvalid ? e[c * 4 + 1] * inv : 0.f;
        o[c].z = rowvalid ? e[c * 4 + 2] * inv : 0.f;
        o[c].w = rowvalid ? e[c * 4 + 3] * inv : 0.f;
    }
#pragma unroll
    for (int c = 0; c < 4; ++c)
        reinterpret_cast<float4*>(drow)[c * 32 + lane] = o[c];
}

// ---------------------------------------------------------------------------
// Column softmax (over agents axis, stride M): one block per (batch, 32-col
// tile). Tile staged in padded LDS; column sums of exp computed with
// V_WMMA_F32_16X16X4_F32: D = ones(16x4) x B(4x16) + C accumulates exact fp32
// column sums (every row of D equals the column sums; invariant to the K-row
// striping permutation because A is all-ones).
// In-place safe: the block loads its whole tile (barrier) before storing.
// ---------------------------------------------------------------------------
__global__ __launch_bounds__(256) void col_softmax_kernel(
    float* __restrict__ x,
    const int* __restrict__ fa_num, const int* __restrict__ tn_num)
{
    extern __shared__ float smem[];
    float* tile = smem;                      // 512 * 33
    float* pmax = tile + NROW * TILE_STRIDE; // 256
    float* cmax = pmax + 256;                // 32
    float* psum = cmax + 32;                 // 256

    const int t  = threadIdx.x;
    const int tx = t & 31;        // column within tile (== lane)
    const int ty = t >> 5;        // wave id (8 waves)
    const int b  = blockIdx.y;
    const int col0 = blockIdx.x * 32;
    const int fa = fa_num[b];
    const int tn = tn_num[b];

    float* base = x + ((size_t)b << 18) + col0;   // b*512*512 + col0

    // Load tile: per row-step each wave loads 32 consecutive floats (128B).
    for (int n = ty; n < NROW; n += 8)
        tile[n * TILE_STRIDE + tx] = base[(size_t)n * MCOL + tx];
    __syncthreads();

    if (fa == 0 || tn == 0) {                      // block-uniform: all masked
        for (int n = ty; n < NROW; n += 8)
            base[(size_t)n * MCOL + tx] = 0.f;
        return;
    }

    // Per-column max over valid rows n < fa.
    float pm = -INFINITY;
    for (int n = ty; n < fa; n += 8)
        pm = fmaxf(pm, tile[n * TILE_STRIDE + tx]);
    pmax[ty * 32 + tx] = pm;
    __syncthreads();
    if (ty == 0) {
        float mm = pmax[tx];
#pragma unroll
        for (int k = 1; k < 8; ++k) mm = fmaxf(mm, pmax[k * 32 + tx]);
        cmax[tx] = mm;
    }
    __syncthreads();

    // exp + WMMA column sums. Wave ty owns rows [ty*64, ty*64+64).
    // Per K-step: B holds 4 rows x 16 cols; lanes 0-15 supply rows (r0,r0+1),
    // lanes 16-31 rows (r0+2,r0+3), column = lane&15. Two col-halves.
    const int lane = tx;
    const int colL = lane & 15;
    const float mL = cmax[colL];
    const float mH = cmax[16 + colL];
    const int rsel = (lane >> 4) << 1;      // 0 or 2

    v8f accLo = {0.f, 0.f, 0.f, 0.f, 0.f, 0.f, 0.f, 0.f};
    v8f accHi = {0.f, 0.f, 0.f, 0.f, 0.f, 0.f, 0.f, 0.f};
    v2f ones;  ones[0] = 1.f; ones[1] = 1.f;

    for (int s = 0; s < 16; ++s) {
        const int r = ty * 64 + s * 4 + rsel;
        const float x0L = tile[(r    ) * TILE_STRIDE + colL];
        const float x1L = tile[(r + 1) * TILE_STRIDE + colL];
        const float x0H = tile[(r    ) * TILE_STRIDE + 16 + colL];
        const float x1H = tile[(r + 1) * TILE_STRIDE + 16 + colL];
        const float e0L = (r     < fa) ? __expf(x0L - mL) : 0.f;
        const float e1L = (r + 1 < fa) ? __expf(x1L - mL) : 0.f;
        const float e0H = (r     < fa) ? __expf(x0H - mH) : 0.f;
        const float e1H = (r + 1 < fa) ? __expf(x1H - mH) : 0.f;
        tile[(r    ) * TILE_STRIDE + colL]      = e0L;
        tile[(r + 1) * TILE_STRIDE + colL]      = e1L;
        tile[(r    ) * TILE_STRIDE + 16 + colL] = e0H;
        tile[(r + 1) * TILE_STRIDE + 16 + colL] = e1H;

        v2f bLo;  bLo[0] = e0L;  bLo[1] = e1L;
        v2f bHi;  bHi[0] = e0H;  bHi[1] = e1H;
        // D = ones(16x4) x B(4x16) + C  -> every D row = column sums of B.
        accLo = __builtin_amdgcn_wmma_f32_16x16x4_f32(
            false, ones, false, bLo, (short)0, accLo, false, false);
        accHi = __builtin_amdgcn_wmma_f32_16x16x4_f32(
            false, ones, false, bHi, (short)0, accHi, false, false);
    }

    // D VGPR0, lanes 0-15: (M=0, N=lane) = colsum[lane] of this wave's rows.
    if (lane < 16) {
        psum[ty * 32 + lane]      = accLo[0];
        psum[ty * 32 + 16 + lane] = accHi[0];
    }
    __syncthreads();

    float sum = 0.f;
#pragma unroll
    for (int k = 0; k < 8; ++k) sum += psum[k * 32 + tx];
    const float inv = 1.f / sum;            // sum >= 1 for valid columns
    const bool colvalid = (col0 + tx) < tn;

    for (int n = ty; n < NROW; n += 8) {
        const float outv = (colvalid && n < fa)
                               ? tile[n * TILE_STRIDE + tx] * inv
                               : 0.f;
        base[(size_t)n * MCOL + tx] = outv;
    }
}

// ---------------------------------------------------------------------------
extern "C" void kernel_launch(void* const* d_in, const int* in_sizes, int n_in,
                              void* d_out, int out_size, void* d_ws, size_t ws_size,
                              hipStream_t stream) {
    const float* logits = (const float*)d_in[0];
    const int*   fa     = (const int*)d_in[1];
    const int*   tn     = (const int*)d_in[2];
    float*       out    = (float*)d_out;

    const float invTau = 1.0f;   // 1 / TAU, applied only on the first pass
    const int rowBlocks = (128 * NROW) / 8;             // one wave per row
    const dim3 colGrid(MCOL / 32, 128);
    const size_t ldsBytes =
        (NROW * TILE_STRIDE + 256 + 32 + 256) * sizeof(float);

    // Iteration 1: row softmax reads logits (applies 1/TAU), then all passes
    // run in place on d_out.
    row_softmax_kernel<<<rowBlocks, 256, 0, stream>>>(logits, out, fa, tn, invTau);
    col_softmax_kernel<<<colGrid, 256, ldsBytes, stream>>>(out, fa, tn);
    for (int it = 1; it < 5; ++it) {
        row_softmax_kernel<<<rowBlocks, 256, 0, stream>>>(out, out, fa, tn, 1.0f);
        col_softmax_kernel<<<colGrid, 256, ldsBytes, stream>>>(out, fa, tn);
    }
}